// EHawkesProcess_6442450944056
// MI455X (gfx1250) — compile-verified
//
#include <hip/hip_runtime.h>
#include <hip/hip_bf16.h>
#include <math.h>

// Fixed problem shape from the reference: (B, S, NT, E) = (8, 256, 50, 64)
#define BB  8
#define SS  256
#define NTT 50
#define EE  64

typedef __attribute__((ext_vector_type(2))) float v2f;
typedef __attribute__((ext_vector_type(8))) float v8f;

__device__ __forceinline__ float softplus_f(float x) {
  // logaddexp(x, 0) = max(x,0) + log1p(exp(-|x|))  (matches jax.nn.softplus, stable)
  return fmaxf(x, 0.0f) + log1pf(expf(-fabsf(x)));
}

// Sum 64 f32 values held 2-per-lane across a wave32 using V_WMMA_F32_16X16X4_F32.
// A-matrix layout (16x4 f32): lanes 0-15 supply {K0,K1}, lanes 16-31 supply {K2,K3},
// row M = lane & 15. With B = ones, D[m,n] = row-sum of A. Summing the 8 D VGPRs
// gives rows 0-7 (lanes<16) / rows 8-15 (lanes>=16); one xor-16 shuffle completes it.
// Placement-agnostic: result is the sum of all 64 inputs regardless of lane mapping.
__device__ __forceinline__ float wave_sum64(float s_lo, float s_hi) {
  v2f amat; amat[0] = s_lo; amat[1] = s_hi;
  v2f bmat; bmat[0] = 1.0f; bmat[1] = 1.0f;
  v8f cacc = {};
  cacc = __builtin_amdgcn_wmma_f32_16x16x4_f32(
      /*neg_a=*/false, amat, /*neg_b=*/false, bmat,
      /*c_mod=*/(short)0, cacc, /*reuse_a=*/false, /*reuse_b=*/false);
  float r = cacc[0] + cacc[1] + cacc[2] + cacc[3] +
            cacc[4] + cacc[5] + cacc[6] + cacc[7];
  r += __shfl_xor(r, 16, 32);
  return r;  // full sum in every lane
}

// One wave32 per (b, i): each lane owns the contiguous element pair e = {2*lane, 2*lane+1}
// so A/P/emb row accesses are single global_load_b64 per row. Accumulate excitation
// m[e] over history j < i, then ll_parts[b*S+i] = valid ? log(sum_e softplus(...) + eps) : 0
__global__ __launch_bounds__(32)
void hawkes_ll_kernel(const float* __restrict__ times, const int* __restrict__ types,
                      const float* __restrict__ emb, const float* __restrict__ a,
                      const float* __restrict__ Amat, const float* __restrict__ Pmat,
                      float* __restrict__ ll_parts) {
  const int bi   = blockIdx.x;          // 0 .. B*S-1
  const int b    = bi / SS;
  const int i    = bi % SS;
  const int lane = threadIdx.x;         // 0..31

  const int   ti  = types[b * SS + i];
  const float t_i = times[b * SS + i];
  const float2 ei = ((const float2*)(emb + ti * EE))[lane];

  float m0 = 0.0f, m1 = 0.0f;
  for (int j = 0; j < i; ++j) {
    const int   tj = types[b * SS + j];
    const float dt = t_i - times[b * SS + j];
    const float2 ar = ((const float2*)(Amat + (tj * NTT + ti) * EE))[lane];
    const float2 pr = ((const float2*)(Pmat + (tj * NTT + ti) * EE))[lane];
    const float2 ej = ((const float2*)(emb + tj * EE))[lane];
    // contrib = emb_i*A*exp(-(emb_i*P)*emb_j*dt)*emb_j = g*A*exp(-g*P*dt), g = emb_i*emb_j
    const float g0 = ei.x * ej.x;
    const float g1 = ei.y * ej.y;
    m0 += g0 * ar.x * expf(-g0 * pr.x * dt);
    m1 += g1 * ar.y * expf(-g1 * pr.y * dt);
  }

  const float2 av = ((const float2*)(a + ti * EE))[lane];
  const float x0 = ei.x * av.x + m0;
  const float x1 = ei.y * av.y + m1;
  const float inten = wave_sum64(softplus_f(x0), softplus_f(x1));

  if (lane == 0) {
    ll_parts[bi] = (t_i >= 0.0f) ? logf(inten + 1e-16f) : 0.0f;
  }
}

// One wave32 per (b, k): m_T[b,k,e] = sum_i valid g*A[ti,k,e]*exp(-g*P[ti,k,e]*dT),
// g = emb[k,e]*emb_t[b,i,e]; then intenT_parts[b*NT+k] = sum_e softplus(emb*a + m_T)
__global__ __launch_bounds__(32)
void hawkes_integ_kernel(const float* __restrict__ times, const int* __restrict__ types,
                         const float* __restrict__ emb, const float* __restrict__ a,
                         const float* __restrict__ Amat, const float* __restrict__ Pmat,
                         const int* __restrict__ Tptr, float* __restrict__ intenT_parts) {
  const int bk   = blockIdx.x;          // 0 .. B*NT-1
  const int b    = bk / NTT;
  const int k    = bk % NTT;
  const int lane = threadIdx.x;

  const float Tf  = (float)(*Tptr);
  const float2 ek = ((const float2*)(emb + k * EE))[lane];

  float m0 = 0.0f, m1 = 0.0f;
  for (int i = 0; i < SS; ++i) {
    const float t = times[b * SS + i];   // wave-uniform -> scalar branch, EXEC stays full
    if (t >= 0.0f) {
      const int   ti = types[b * SS + i];
      const float dT = Tf - t;
      const float2 ar = ((const float2*)(Amat + (ti * NTT + k) * EE))[lane];
      const float2 pr = ((const float2*)(Pmat + (ti * NTT + k) * EE))[lane];
      const float2 et = ((const float2*)(emb + ti * EE))[lane];
      const float g0 = ek.x * et.x;
      const float g1 = ek.y * et.y;
      m0 += g0 * ar.x * expf(-g0 * pr.x * dT);
      m1 += g1 * ar.y * expf(-g1 * pr.y * dT);
    }
  }

  const float2 av = ((const float2*)(a + k * EE))[lane];
  const float x0 = ek.x * av.x + m0;
  const float x1 = ek.y * av.y + m1;
  const float s  = wave_sum64(softplus_f(x0), softplus_f(x1));

  if (lane == 0) intenT_parts[bk] = s;
}

// Single block: deterministic fixed-order reductions + final combine.
__global__ __launch_bounds__(256)
void hawkes_final_kernel(const float* __restrict__ times, const float* __restrict__ emb,
                         const float* __restrict__ a, const int* __restrict__ Tptr,
                         const float* __restrict__ ll_parts,
                         const float* __restrict__ intenT_parts,
                         float* __restrict__ out) {
  __shared__ float red[256];
  const int t = threadIdx.x;
  const float Tf = (float)(*Tptr);

  // 1) ll = sum of per-event log intensities
  float s = 0.0f;
  for (int idx = t; idx < BB * SS; idx += 256) s += ll_parts[idx];
  red[t] = s; __syncthreads();
  for (int off = 128; off > 0; off >>= 1) { if (t < off) red[t] += red[t + off]; __syncthreads(); }
  const float ll = red[0]; __syncthreads();

  // 2) base_sum = sum softplus(emb * a) over NT*E
  s = 0.0f;
  for (int idx = t; idx < NTT * EE; idx += 256) s += softplus_f(emb[idx] * a[idx]);
  red[t] = s; __syncthreads();
  for (int off = 128; off > 0; off >>= 1) { if (t < off) red[t] += red[t + off]; __syncthreads(); }
  const float base_sum = red[0]; __syncthreads();

  // 3) per-batch integral
  float integral = 0.0f;
  for (int b = 0; b < BB; ++b) {
    red[t] = (t < NTT) ? intenT_parts[b * NTT + t] : 0.0f; __syncthreads();
    for (int off = 128; off > 0; off >>= 1) { if (t < off) red[t] += red[t + off]; __syncthreads(); }
    const float intenT = red[0]; __syncthreads();

    const float tm = times[b * SS + t];     // S == 256 == blockDim
    const bool  v  = (tm >= 0.0f);

    red[t] = v ? tm : -1e30f; __syncthreads();
    for (int off = 128; off > 0; off >>= 1) { if (t < off) red[t] = fmaxf(red[t], red[t + off]); __syncthreads(); }
    const float last = red[0]; __syncthreads();

    red[t] = v ? tm : 1e30f; __syncthreads();
    for (int off = 128; off > 0; off >>= 1) { if (t < off) red[t] = fminf(red[t], red[t + off]); __syncthreads(); }
    const float first = red[0]; __syncthreads();

    red[t] = v ? 1.0f : 0.0f; __syncthreads();
    for (int off = 128; off > 0; off >>= 1) { if (t < off) red[t] += red[t + off]; __syncthreads(); }
    const bool any_v = (red[0] > 0.0f); __syncthreads();

    integral += any_v ? (intenT * (Tf - last) + base_sum * first)
                      : (base_sum * Tf);
  }

  if (t == 0) out[0] = -(ll - integral);
}

extern "C" void kernel_launch(void* const* d_in, const int* in_sizes, int n_in,
                              void* d_out, int out_size, void* d_ws, size_t ws_size,
                              hipStream_t stream) {
  // Inputs in setup_inputs() order:
  const float* times = (const float*)d_in[0];   // (B,S) f32
  const int*   types = (const int*)  d_in[1];   // (B,S) i32
  const int*   Tptr  = (const int*)  d_in[2];   // scalar int T
  const float* emb   = (const float*)d_in[3];   // (NT,E) f32
  const float* a     = (const float*)d_in[4];   // (NT,E) f32
  const float* Amat  = (const float*)d_in[5];   // (NT,NT,E) f32
  const float* Pmat  = (const float*)d_in[6];   // (NT,NT,E) f32
  float* out = (float*)d_out;

  float* ll_parts     = (float*)d_ws;           // B*S   = 2048 floats
  float* intenT_parts = ll_parts + BB * SS;     // B*NT  = 400 floats

  hipLaunchKernelGGL(hawkes_ll_kernel, dim3(BB * SS), dim3(32), 0, stream,
                     times, types, emb, a, Amat, Pmat, ll_parts);
  hipLaunchKernelGGL(hawkes_integ_kernel, dim3(BB * NTT), dim3(32), 0, stream,
                     times, types, emb, a, Amat, Pmat, Tptr, intenT_parts);
  hipLaunchKernelGGL(hawkes_final_kernel, dim3(1), dim3(256), 0, stream,
                     times, emb, a, Tptr, ll_parts, intenT_parts, out);
}